// BERT_BiLSTM_CRF_77541339562223
// MI455X (gfx1250) — compile-verified
//
#include <hip/hip_runtime.h>
#include <hip/hip_bf16.h>

typedef __attribute__((ext_vector_type(2))) float v2f;
typedef __attribute__((ext_vector_type(8))) float v8f;

#define GD 1024           // feature dim D
#define TILE_M 128
#define TILE_N 128
#define TILE_K 32
#define LDA 36            // padded LDS stride for A (floats); 36*4=144 -> 16B aligned rows
#define LDB 132           // padded LDS stride for B (floats); 132*4=528 -> 16B aligned rows

// ---------------- zero workspace ----------------
__global__ void k_zero(float4* __restrict__ p, int n4) {
    int i = blockIdx.x * blockDim.x + threadIdx.x;
    if (i < n4) p[i] = make_float4(0.f, 0.f, 0.f, 0.f);
}

// ---------------- per-edge degree/weight accumulation ----------------
__global__ void k_edge_deg(const float* __restrict__ ew,
                           const int* __restrict__ src, const int* __restrict__ dst,
                           float* __restrict__ w_out, float* __restrict__ w_in,
                           float* __restrict__ d_outd, float* __restrict__ d_ind, int E) {
    int e = blockIdx.x * blockDim.x + threadIdx.x;
    if (e >= E) return;
    float w = ew[e];
    int s = src[e], d = dst[e];
    atomicAdd(w_out + s, w);
    atomicAdd(w_in + d, w);
    atomicAdd(d_outd + s, 1.0f);
    atomicAdd(d_ind + d, 1.0f);
}

// ---------------- per-node factors ----------------
__global__ void k_node(const float* __restrict__ w_out, const float* __restrict__ w_in,
                       const float* __restrict__ d_outd, const float* __restrict__ d_ind,
                       float* __restrict__ so, float* __restrict__ si,
                       float* __restrict__ rin, int N) {
    int i = blockIdx.x * blockDim.x + threadIdx.x;
    if (i >= N) return;
    so[i]  = rsqrtf(w_out[i]) * rsqrtf(fmaxf(d_outd[i], 1.0f));
    si[i]  = rsqrtf(w_in[i]);
    rin[i] = rsqrtf(fmaxf(d_ind[i], 1.0f));
}

// ---------------- per-edge coefficient ----------------
__global__ void k_coef(const float* __restrict__ ew,
                       const int* __restrict__ src, const int* __restrict__ dst,
                       const float* __restrict__ so, const float* __restrict__ si,
                       float* __restrict__ coef, int E) {
    int e = blockIdx.x * blockDim.x + threadIdx.x;
    if (e >= E) return;
    coef[e] = ew[e] * so[src[e]] * si[dst[e]];
}

// ---------------- scatter: agg[dst] += coef * x[src], one block per edge ----------------
__global__ __launch_bounds__(256)
void k_scatter(const float* __restrict__ x, const int* __restrict__ src,
               const int* __restrict__ dst, const float* __restrict__ coef,
               float* __restrict__ agg) {
    int e = blockIdx.x;
    float c = coef[e];
    size_t sb = (size_t)src[e] * GD;
    size_t db = (size_t)dst[e] * GD;
    int j = threadIdx.x * 4;
    float4 xv = *(const float4*)(x + sb + j);
    atomicAdd(agg + db + j + 0, c * xv.x);
    atomicAdd(agg + db + j + 1, c * xv.y);
    atomicAdd(agg + db + j + 2, c * xv.z);
    atomicAdd(agg + db + j + 3, c * xv.w);
}

// ---------------- WMMA f32 GEMM: out = relu(diag(rin) * (agg @ W) + b) ----------------
// Double-buffered async (memory -> LDS) pipeline: copy of tile t+1 overlaps WMMA on tile t.
__global__ __launch_bounds__(256)
void k_gemm(const float* __restrict__ A,    // agg [N x D]
            const float* __restrict__ rin,  // [N]
            const float* __restrict__ W,    // [D x D]
            const float* __restrict__ bias, // [D]
            float* __restrict__ out) {      // [N x D]
    __shared__ float As[2][TILE_M * LDA];
    __shared__ float Bs[2][TILE_K * LDB];

    const int nColBlocks = GD / TILE_N;           // 8
    int bx = blockIdx.x % nColBlocks;
    int by = blockIdx.x / nColBlocks;
    int rowBase = by * TILE_M;
    int colBase = bx * TILE_N;

    int tid  = threadIdx.x;
    int wid  = tid >> 5;                          // wave 0..7
    int lane = tid & 31;
    int ln   = lane & 15;
    int half = lane >> 4;                         // 0 or 1
    int waveRow = (wid >> 2) * 64;                // 0 / 64
    int waveCol = (wid & 3) * 32;                 // 0,32,64,96

    uint32_t asBase[2] = { (uint32_t)(uintptr_t)&As[0][0], (uint32_t)(uintptr_t)&As[1][0] };
    uint32_t bsBase[2] = { (uint32_t)(uintptr_t)&Bs[0][0], (uint32_t)(uintptr_t)&Bs[1][0] };

    // Stage one 128x32 A tile + 32x128 B tile into LDS buffer `buf` with async copies.
    // 8 x b128 async ops per thread, tracked by ASYNCcnt.
    auto stage = [&](int kk, int buf) {
#pragma unroll
        for (int i = 0; i < 4; ++i) {
            int idx = tid + i * 256;              // 0..1023
            int r = idx >> 3;                     // 0..127
            int c = (idx & 7) << 2;               // 0..28
            const float* ga = A + (size_t)(rowBase + r) * GD + kk + c;
            uint32_t la = asBase[buf] + (uint32_t)((r * LDA + c) * 4);
            asm volatile("global_load_async_to_lds_b128 %0, %1, off"
                         :: "v"(la), "v"(ga) : "memory");
        }
#pragma unroll
        for (int i = 0; i < 4; ++i) {
            int idx = tid + i * 256;
            int r = idx >> 5;                     // 0..31
            int c = (idx & 31) << 2;              // 0..124
            const float* gb = W + (size_t)(kk + r) * GD + colBase + c;
            uint32_t lb = bsBase[buf] + (uint32_t)((r * LDB + c) * 4);
            asm volatile("global_load_async_to_lds_b128 %0, %1, off"
                         :: "v"(lb), "v"(gb) : "memory");
        }
    };

    v8f acc[4][2] = {};

    stage(0, 0);
    const int nk = GD / TILE_K;                   // 32
    for (int t = 0; t < nk; ++t) {
        int cur = t & 1;
        // my stage(t) copies have landed in LDS:
        asm volatile("s_wait_asynccnt 0x0" ::: "memory");
        // everyone's stage(t) copies visible; everyone done reading buf[cur^1] (compute t-1):
        __syncthreads();
        if (t + 1 < nk) stage((t + 1) * TILE_K, (t + 1) & 1);   // overlaps with compute below

        const float* Asb = &As[cur][0];
        const float* Bsb = &Bs[cur][0];
#pragma unroll
        for (int k = 0; k < TILE_K; k += 4) {
            // A fragments: 16x4 f32, lanes 0-15 -> K=k..k+1, lanes 16-31 -> K=k+2..k+3
            v2f aF[4];
#pragma unroll
            for (int rt = 0; rt < 4; ++rt) {
                const float* p = Asb + (waveRow + rt * 16 + ln) * LDA + k + 2 * half;
                aF[rt] = *(const v2f*)p;
            }
            // B fragments: 4x16 f32, VGPR0 -> rows k / k+2, VGPR1 -> rows k+1 / k+3
            v2f bF[2];
#pragma unroll
            for (int ct = 0; ct < 2; ++ct) {
                int col = waveCol + ct * 16 + ln;
                v2f b;
                b.x = Bsb[(k + 2 * half) * LDB + col];
                b.y = Bsb[(k + 1 + 2 * half) * LDB + col];
                bF[ct] = b;
            }
#pragma unroll
            for (int rt = 0; rt < 4; ++rt)
#pragma unroll
                for (int ct = 0; ct < 2; ++ct)
                    acc[rt][ct] = __builtin_amdgcn_wmma_f32_16x16x4_f32(
                        false, aF[rt], false, bF[ct], (short)0, acc[rt][ct],
                        false, false);
        }
    }

    // epilogue: row scale (rsqrt deg_in) + bias + relu
    // C/D layout: VGPR r -> M = r + 8*half, N = ln
    float bv[2];
#pragma unroll
    for (int ct = 0; ct < 2; ++ct)
        bv[ct] = bias[colBase + waveCol + ct * 16 + ln];
#pragma unroll
    for (int rt = 0; rt < 4; ++rt) {
        int r0 = rowBase + waveRow + rt * 16 + half * 8;
#pragma unroll
        for (int r = 0; r < 8; ++r) {
            float rv = rin[r0 + r];
#pragma unroll
            for (int ct = 0; ct < 2; ++ct) {
                int col = colBase + waveCol + ct * 16 + ln;
                float v = fmaf(acc[rt][ct][r], rv, bv[ct]);
                out[(size_t)(r0 + r) * GD + col] = fmaxf(v, 0.0f);
            }
        }
    }
}

extern "C" void kernel_launch(void* const* d_in, const int* in_sizes, int n_in,
                              void* d_out, int out_size, void* d_ws, size_t ws_size,
                              hipStream_t stream) {
    const float* x      = (const float*)d_in[0];   // [N, D]
    const float* edge_w = (const float*)d_in[1];   // [E]
    const float* W      = (const float*)d_in[2];   // [D, D]
    const float* b      = (const float*)d_in[3];   // [D]
    const int*   src    = (const int*)d_in[4];     // [E]
    const int*   dst    = (const int*)d_in[5];     // [E]
    float* out = (float*)d_out;

    const int E = in_sizes[1];                     // 131072
    const int N = in_sizes[0] / GD;                // 16384

    // workspace layout (floats): [agg N*D][w_out N][w_in N][deg_out N][deg_in N][so N][si N][rin N][coef E]
    float* ws      = (float*)d_ws;
    float* agg     = ws;
    float* w_out   = agg + (size_t)N * GD;
    float* w_in    = w_out + N;
    float* deg_out = w_in + N;
    float* deg_in  = deg_out + N;
    float* so      = deg_in + N;
    float* si      = so + N;
    float* rin     = si + N;
    float* coef    = rin + N;

    // zero agg + the 4 accumulation arrays (contiguous region)
    size_t zeroF = (size_t)N * GD + 4 * (size_t)N;
    int n4 = (int)(zeroF / 4);
    k_zero<<<(n4 + 255) / 256, 256, 0, stream>>>((float4*)ws, n4);

    k_edge_deg<<<(E + 255) / 256, 256, 0, stream>>>(edge_w, src, dst,
                                                    w_out, w_in, deg_out, deg_in, E);
    k_node<<<(N + 255) / 256, 256, 0, stream>>>(w_out, w_in, deg_out, deg_in,
                                                so, si, rin, N);
    k_coef<<<(E + 255) / 256, 256, 0, stream>>>(edge_w, src, dst, so, si, coef, E);
    k_scatter<<<E, 256, 0, stream>>>(x, src, dst, coef, agg);
    k_gemm<<<(N / TILE_M) * (GD / TILE_N), 256, 0, stream>>>(agg, rin, W, b, out);
}